// ClickbaitDetectorV5_29386166239844
// MI455X (gfx1250) — compile-verified
//
#include <hip/hip_runtime.h>
#include <hip/hip_bf16.h>

#define BB   128
#define SS   256
#define HH   768
#define KK   6
#define DEXP 512
#define DBASE 3846
#define DIN   3942
#define DIN_PAD  3968   // round up to multiple of 64 (zero padded)
#define DCLS  6918
#define DCLS_PAD 6976   // round up to multiple of 64 (zero padded)

typedef float v2f __attribute__((ext_vector_type(2)));
typedef float v8f __attribute__((ext_vector_type(8)));

// Low 32 bits of a flat pointer to __shared__ == LDS byte offset (ISA 10.2)
__device__ __forceinline__ unsigned lds_off(const void* p) {
    return (unsigned)(unsigned long long)p;
}
__device__ __forceinline__ void async_copy_b128(const float* src, unsigned dst) {
    asm volatile("global_load_async_to_lds_b128 %0, %1, off"
                 :: "v"(dst), "v"(src) : "memory");
}

// ---------------------------------------------------------------------------
// 1) Per-batch segment info: sep1, sep2, vlen, nsep
// ---------------------------------------------------------------------------
__global__ void seg_info_kernel(const int* __restrict__ ids,
                                const int* __restrict__ am,
                                int* __restrict__ seg) {
    int b = blockIdx.x * blockDim.x + threadIdx.x;
    if (b >= BB) return;
    const int* row = ids + (long)b * SS;
    const int* amr = am + (long)b * SS;
    int csum = 0, sep1 = SS, sep2 = SS, vlen = 0;
    for (int s = 0; s < SS; ++s) {
        if (row[s] == 2) {
            ++csum;
            if (csum == 1 && sep1 == SS) sep1 = s;
            if (csum == 2 && sep2 == SS) sep2 = s;
        }
        vlen += amr[s];
    }
    seg[b * 4 + 0] = sep1;
    seg[b * 4 + 1] = sep2;
    seg[b * 4 + 2] = vlen;
    seg[b * 4 + 3] = csum;
}

// ---------------------------------------------------------------------------
// 2) Fused layer-softmax weighted sum + masked segment mean pool.
//    Only rows inside a segment mask are loaded from the 403MB hidden4
//    tensor (bandwidth-dominant stage).  Writes z_base into d_out.
// ---------------------------------------------------------------------------
__global__ __launch_bounds__(256)
void pool_kernel(const float* __restrict__ hidden4,
                 const float* __restrict__ last_hidden,
                 const float* __restrict__ layer_w,
                 const float* __restrict__ pt,
                 const int* __restrict__ seg,
                 float* __restrict__ zb) {
    const long LSTRIDE = (long)BB * SS * HH;
    int b = blockIdx.y;
    int h = blockIdx.x * 256 + threadIdx.x;  // < 768

    float w0 = layer_w[0], w1 = layer_w[1], w2 = layer_w[2], w3 = layer_w[3];
    float m = fmaxf(fmaxf(w0, w1), fmaxf(w2, w3));
    float e0 = expf(w0 - m), e1 = expf(w1 - m), e2 = expf(w2 - m), e3 = expf(w3 - m);
    float inv = 1.0f / (e0 + e1 + e2 + e3);
    e0 *= inv; e1 *= inv; e2 *= inv; e3 *= inv;

    int sep1 = seg[b * 4 + 0], sep2 = seg[b * 4 + 1];
    int vlen = seg[b * 4 + 2], nsep = seg[b * 4 + 3];
    bool has2 = nsep >= 2;
    int endp = min(vlen - 1, SS);

    float au = 0.f, av = 0.f;
    int cu = 0, cv = 0;
    const long base = (long)b * SS * HH + h;
    for (int s = 0; s < SS; ++s) {
        bool tm, lm;
        if (has2) {
            tm = (s >= 1) && (s < sep1);
            lm = (s >= sep2 + 1) && (s < endp);
        } else {
            bool fb = (vlen > 2) ? ((s >= 1) && (s < vlen - 1)) : (s == 0);
            tm = fb; lm = fb;
        }
        if (tm || lm) {
            long o = base + (long)s * HH;
            float x = e0 * hidden4[o] + e1 * hidden4[o + LSTRIDE] +
                      e2 * hidden4[o + 2 * LSTRIDE] + e3 * hidden4[o + 3 * LSTRIDE];
            if (tm) { au += x; ++cu; }
            if (lm) { av += x; ++cv; }
        }
    }
    float u = (cu > 0) ? au / (float)cu : 0.f;
    float v = (cv > 0) ? av / (float)cv : 0.f;

    float* zrow = zb + (long)b * DBASE;
    zrow[h]        = last_hidden[(long)b * SS * HH + h];  // cls token
    zrow[768 + h]  = u;
    zrow[1536 + h] = v;
    zrow[2304 + h] = fabsf(u - v);
    zrow[3072 + h] = u * v;
    if (blockIdx.x == 0 && threadIdx.x < 6)
        zrow[3840 + threadIdx.x] = pt[b * 6 + threadIdx.x];
}

// ---------------------------------------------------------------------------
// 3) K-head attention pooling over last_hidden -> pooled (K, B, H)
// ---------------------------------------------------------------------------
__global__ __launch_bounds__(256)
void attn_kernel(const float* __restrict__ lh,
                 const int* __restrict__ am,
                 const float* __restrict__ attn_w,
                 const float* __restrict__ attn_b,
                 float* __restrict__ pooled) {
    __shared__ float sw[KK * HH];
    __shared__ float sc[KK][SS];
    int b = blockIdx.x;
    int tid = threadIdx.x;
    for (int i = tid; i < KK * HH; i += 256) sw[i] = attn_w[i];
    __syncthreads();

    int wave = tid >> 5, lane = tid & 31;
    const float* lhb = lh + (long)b * SS * HH;

    for (int si = 0; si < 32; ++si) {
        int s = wave * 32 + si;
        const float* row = lhb + (long)s * HH;
        float a[KK];
#pragma unroll
        for (int k = 0; k < KK; ++k) a[k] = 0.f;
        for (int hh = lane; hh < HH; hh += 32) {
            float x = row[hh];
#pragma unroll
            for (int k = 0; k < KK; ++k) a[k] += x * sw[k * HH + hh];
        }
#pragma unroll
        for (int k = 0; k < KK; ++k)
            for (int mm = 16; mm >= 1; mm >>= 1) a[k] += __shfl_xor(a[k], mm);
        if (lane == 0) {
            int msk = am[(long)b * SS + s];
#pragma unroll
            for (int k = 0; k < KK; ++k)
                sc[k][s] = msk ? (a[k] + attn_b[k]) : -10000.0f;
        }
    }
    __syncthreads();

    if (wave < KK) {
        int k = wave;
        float mx = -1e30f;
        for (int s = lane; s < SS; s += 32) mx = fmaxf(mx, sc[k][s]);
        for (int mm = 16; mm >= 1; mm >>= 1) mx = fmaxf(mx, __shfl_xor(mx, mm));
        float sum = 0.f;
        for (int s = lane; s < SS; s += 32) {
            float e = expf(sc[k][s] - mx);
            sc[k][s] = e;
            sum += e;
        }
        for (int mm = 16; mm >= 1; mm >>= 1) sum += __shfl_xor(sum, mm);
        float inv = 1.0f / sum;
        for (int s = lane; s < SS; s += 32) sc[k][s] *= inv;
    }
    __syncthreads();

    for (int h0 = tid; h0 < HH; h0 += 256) {
        float p[KK];
#pragma unroll
        for (int k = 0; k < KK; ++k) p[k] = 0.f;
        for (int s = 0; s < SS; ++s) {
            float x = lhb[(long)s * HH + h0];
#pragma unroll
            for (int k = 0; k < KK; ++k) p[k] += sc[k][s] * x;
        }
#pragma unroll
        for (int k = 0; k < KK; ++k)
            pooled[((long)k * BB + b) * HH + h0] = p[k];
    }
}

// ---------------------------------------------------------------------------
// 4) Batched GEMM: V_WMMA_F32_16X16X4_F32 + double-buffered
//    GLOBAL_LOAD_ASYNC_TO_LDS_B128 tile pipeline (ASYNCcnt tracked).
//    C[M,N] = act(A[M,Kd](stride lda) * W[Kd,N] + bias)
//    Requirements: M %32==0, N %64==0, lda %64==0 with A zero-padded
//    in columns [Kd, lda).  W ragged tail handled by a guarded sync tile.
// ---------------------------------------------------------------------------
__global__ __launch_bounds__(256)
void gemm_wmma_kernel(const float* __restrict__ A, const float* __restrict__ W,
                      const float* __restrict__ bias, float* __restrict__ C,
                      int M, int N, int Kd, int lda,
                      long sA, long sW, long sBias, long sC, int act) {
    __shared__ float As[2][32][68];   // row stride 272B: 16B aligned, no bank conflicts
    __shared__ float Ws[2][64][68];
    const float* Ab = A + (long)blockIdx.z * sA;
    const float* Wb = W + (long)blockIdx.z * sW;
    float* Cb = C + (long)blockIdx.z * sC;

    int tid = threadIdx.x;
    int wave = tid >> 5, lane = tid & 31;
    int wm = wave >> 2, wn = wave & 3;  // 2x4 wave grid -> 32x64 block tile
    int rowBase = blockIdx.y * 32;
    int colBase = blockIdx.x * 64;

    // --- tile stagers -----------------------------------------------------
    auto issueA = [&](int kb, int bf) {           // 2 async b128 per thread
#pragma unroll
        for (int i = 0; i < 2; ++i) {
            int idx = tid + i * 256;
            int r = idx >> 4, c = (idx & 15) << 2;
            const float* src = Ab + (long)(rowBase + r) * lda + kb + c;
            async_copy_b128(src, lds_off(&As[bf][r][c]));
        }
    };
    auto issueWAsync = [&](int kb, int bf) {      // 4 async b128 per thread
#pragma unroll
        for (int i = 0; i < 4; ++i) {
            int idx = tid + i * 256;
            int r = idx >> 4, c = (idx & 15) << 2;
            const float* src = Wb + (long)(kb + r) * N + colBase + c;
            async_copy_b128(src, lds_off(&Ws[bf][r][c]));
        }
    };
    auto issueWSync = [&](int kb, int bf) {       // guarded tail tile
#pragma unroll
        for (int i = 0; i < 16; ++i) {
            int idx = tid + i * 256;
            int r = idx >> 6, c = idx & 63;
            int k = kb + r;
            Ws[bf][r][c] = (k < Kd) ? Wb[(long)k * N + colBase + c] : 0.f;
        }
    };

    v8f acc = {0.f, 0.f, 0.f, 0.f, 0.f, 0.f, 0.f, 0.f};
    const int KT = (Kd + 63) >> 6;   // total k-tiles
    const int F  = Kd >> 6;          // full (async-eligible) W tiles

    // prologue: stage tile 0 into buffer 0
    issueA(0, 0);
    if (F > 0) issueWAsync(0, 0); else issueWSync(0, 0);

    int buf = 0;
    for (int t = 0; t < KT; ++t) {
        // stage tile t+1 into the other buffer, then wait for tile t only
        int nextKind = (t + 1 < KT) ? ((t + 1 < F) ? 2 : 1) : 0;
        if (nextKind == 2) {
            issueA((t + 1) << 6, buf ^ 1);
            issueWAsync((t + 1) << 6, buf ^ 1);
            asm volatile("s_wait_asynccnt 0x6" ::: "memory");
        } else if (nextKind == 1) {
            issueA((t + 1) << 6, buf ^ 1);
            issueWSync((t + 1) << 6, buf ^ 1);
            asm volatile("s_wait_asynccnt 0x2" ::: "memory");
        } else {
            asm volatile("s_wait_asynccnt 0x0" ::: "memory");
        }
        __syncthreads();

        int arow = wm * 16 + (lane & 15);
        int bcol = wn * 16 + (lane & 15);
        int koff = (lane < 16) ? 0 : 2;   // ISA 16x4 f32 A/B lane layout
#pragma unroll
        for (int k4 = 0; k4 < 64; k4 += 4) {
            v2f a, bfv;
            a.x   = As[buf][arow][k4 + koff];
            a.y   = As[buf][arow][k4 + koff + 1];
            bfv.x = Ws[buf][k4 + koff][bcol];
            bfv.y = Ws[buf][k4 + koff + 1][bcol];
            acc = __builtin_amdgcn_wmma_f32_16x16x4_f32(
                false, a, false, bfv, (short)0, acc, false, false);
        }
        __syncthreads();
        buf ^= 1;
    }

    int colN = colBase + wn * 16 + (lane & 15);
    float bval = bias ? bias[(long)blockIdx.z * sBias + colN] : 0.f;
#pragma unroll
    for (int r = 0; r < 8; ++r) {
        int row = rowBase + wm * 16 + ((lane < 16) ? r : r + 8);
        float v = acc[r] + bval;
        if (act == 1) v = 0.5f * v * (1.0f + erff(v * 0.70710678118654752f));
        Cb[(long)row * N + colN] = v;
    }
}

// ---------------------------------------------------------------------------
// 5) In-place LayerNorm over D elems of each row (row stride ldx).
// ---------------------------------------------------------------------------
__global__ __launch_bounds__(256)
void ln_kernel(float* __restrict__ X, const float* __restrict__ g,
               const float* __restrict__ bt, int D, int ldx, int rows_per_batch) {
    __shared__ float red[256];
    __shared__ float red2[256];
    int row = blockIdx.x, tid = threadIdx.x;
    float* x = X + (long)row * ldx;
    float s = 0.f, s2 = 0.f;
    for (int i = tid; i < D; i += 256) { float v = x[i]; s += v; s2 += v * v; }
    red[tid] = s; red2[tid] = s2;
    __syncthreads();
    for (int st = 128; st > 0; st >>= 1) {
        if (tid < st) { red[tid] += red[tid + st]; red2[tid] += red2[tid + st]; }
        __syncthreads();
    }
    float mean = red[0] / (float)D;
    float var = red2[0] / (float)D - mean * mean;
    float inv = rsqrtf(var + 1e-5f);
    long goff = (long)(row / rows_per_batch) * D;
    for (int i = tid; i < D; i += 256)
        x[i] = (x[i] - mean) * inv * g[goff + i] + bt[goff + i];
}

// ---------------------------------------------------------------------------
// 6) expert_aux[b,k] = dot(Z[k,b,:], aux_w[k]) + aux_b[k]   (one wave each)
// ---------------------------------------------------------------------------
__global__ void aux_kernel(const float* __restrict__ Z, const float* __restrict__ aux_w,
                           const float* __restrict__ aux_b, float* __restrict__ out) {
    int idx = blockIdx.x;
    int b = idx / KK, k = idx % KK;
    int lane = threadIdx.x;
    const float* z = Z + ((long)k * BB + b) * DEXP;
    const float* w = aux_w + (long)k * DEXP;
    float acc = 0.f;
    for (int d = lane; d < DEXP; d += 32) acc += z[d] * w[d];
    for (int mm = 16; mm >= 1; mm >>= 1) acc += __shfl_xor(acc, mm);
    if (lane == 0) out[b * KK + k] = acc + aux_b[k];
}

// ---------------------------------------------------------------------------
// 7) Router input r_in = [z_base, cat_emb, src_emb, pt@pat_w+pat_b, 0-pad]
// ---------------------------------------------------------------------------
__global__ __launch_bounds__(256)
void rin_kernel(const float* __restrict__ zb, const float* __restrict__ cat_emb,
                const float* __restrict__ src_emb, const float* __restrict__ pat_w,
                const float* __restrict__ pat_b, const float* __restrict__ pt,
                const int* __restrict__ cat, const int* __restrict__ src,
                float* __restrict__ rin) {
    int b = blockIdx.x, tid = threadIdx.x;
    const float* z = zb + (long)b * DBASE;
    float* r = rin + (long)b * DIN_PAD;
    for (int i = tid; i < DBASE; i += 256) r[i] = z[i];
    if (tid < 32) {
        r[DBASE + tid]      = cat_emb[(long)cat[b] * 32 + tid];
        r[DBASE + 32 + tid] = src_emb[(long)src[b] * 32 + tid];
        float acc = pat_b[tid];
        for (int j = 0; j < 6; ++j) acc += pt[b * 6 + j] * pat_w[j * 32 + tid];
        r[DBASE + 64 + tid] = acc;
    }
    for (int i = DIN + tid; i < DIN_PAD; i += 256) r[i] = 0.f;  // zero pad
}

// ---------------------------------------------------------------------------
// 8) Router head: logits = rh @ r_w2 + r_b2 ; softmax  (one wave per b)
// ---------------------------------------------------------------------------
__global__ void router_kernel(const float* __restrict__ rh, const float* __restrict__ r_w2,
                              const float* __restrict__ r_b2,
                              float* __restrict__ out_rw, float* __restrict__ out_rl) {
    int b = blockIdx.x, lane = threadIdx.x;
    const float* r = rh + (long)b * 512;
    float acc[KK];
#pragma unroll
    for (int k = 0; k < KK; ++k) acc[k] = 0.f;
    for (int d = lane; d < 512; d += 32) {
        float x = r[d];
#pragma unroll
        for (int k = 0; k < KK; ++k) acc[k] += x * r_w2[d * KK + k];
    }
#pragma unroll
    for (int k = 0; k < KK; ++k)
        for (int mm = 16; mm >= 1; mm >>= 1) acc[k] += __shfl_xor(acc[k], mm);
    if (lane == 0) {
        float lg[KK], e[KK];
        float mx = -1e30f;
        for (int k = 0; k < KK; ++k) { lg[k] = acc[k] + r_b2[k]; mx = fmaxf(mx, lg[k]); }
        float sum = 0.f;
        for (int k = 0; k < KK; ++k) { e[k] = expf(lg[k] - mx); sum += e[k]; }
        float inv = 1.0f / sum;
        for (int k = 0; k < KK; ++k) {
            out_rl[b * KK + k] = lg[k];
            out_rw[b * KK + k] = e[k] * inv;
        }
    }
}

// ---------------------------------------------------------------------------
// 9) Classifier input c = [z_base, Z * router_weights, 0-pad] (pre-LN)
// ---------------------------------------------------------------------------
__global__ __launch_bounds__(256)
void cbuild_kernel(const float* __restrict__ zb, const float* __restrict__ Z,
                   const float* __restrict__ rw, float* __restrict__ cin) {
    int b = blockIdx.x, tid = threadIdx.x;
    const float* z = zb + (long)b * DBASE;
    float* c = cin + (long)b * DCLS_PAD;
    for (int i = tid; i < DBASE; i += 256) c[i] = z[i];
    for (int i = tid; i < KK * DEXP; i += 256) {
        int k = i >> 9, d = i & 511;
        c[DBASE + i] = Z[((long)k * BB + b) * DEXP + d] * rw[b * KK + k];
    }
    for (int i = DCLS + tid; i < DCLS_PAD; i += 256) c[i] = 0.f;  // zero pad
}

// ---------------------------------------------------------------------------
// 10) Final logits = c2 @ cls_w3 + cls_b3
// ---------------------------------------------------------------------------
__global__ __launch_bounds__(256)
void final_kernel(const float* __restrict__ c2, const float* __restrict__ w3,
                  const float* __restrict__ b3, float* __restrict__ logits) {
    __shared__ float red[256];
    int b = blockIdx.x, tid = threadIdx.x;
    red[tid] = c2[(long)b * 256 + tid] * w3[tid];
    __syncthreads();
    for (int st = 128; st > 0; st >>= 1) {
        if (tid < st) red[tid] += red[tid + st];
        __syncthreads();
    }
    if (tid == 0) logits[b] = red[0] + b3[0];
}

// ---------------------------------------------------------------------------
extern "C" void kernel_launch(void* const* d_in, const int* in_sizes, int n_in,
                              void* d_out, int out_size, void* d_ws, size_t ws_size,
                              hipStream_t stream) {
    const float* attn_w   = (const float*)d_in[0];
    const float* attn_b   = (const float*)d_in[1];
    const float* ad_w1    = (const float*)d_in[2];
    const float* ad_b1    = (const float*)d_in[3];
    const float* ad_lng   = (const float*)d_in[4];
    const float* ad_lnb   = (const float*)d_in[5];
    const float* ad_w2    = (const float*)d_in[6];
    const float* ad_b2    = (const float*)d_in[7];
    const float* aux_w    = (const float*)d_in[8];
    const float* aux_b    = (const float*)d_in[9];
    const float* cat_emb  = (const float*)d_in[10];
    const float* src_emb  = (const float*)d_in[11];
    const float* pat_w    = (const float*)d_in[12];
    const float* pat_b    = (const float*)d_in[13];
    const float* r_w1     = (const float*)d_in[14];
    const float* r_b1     = (const float*)d_in[15];
    const float* r_lng    = (const float*)d_in[16];
    const float* r_lnb    = (const float*)d_in[17];
    const float* r_w2     = (const float*)d_in[18];
    const float* r_b2     = (const float*)d_in[19];
    const float* cls_lng  = (const float*)d_in[20];
    const float* cls_lnb  = (const float*)d_in[21];
    const float* cls_w1   = (const float*)d_in[22];
    const float* cls_b1   = (const float*)d_in[23];
    const float* cls_w2   = (const float*)d_in[24];
    const float* cls_b2   = (const float*)d_in[25];
    const float* cls_w3   = (const float*)d_in[26];
    const float* cls_b3   = (const float*)d_in[27];
    const float* layer_w  = (const float*)d_in[28];
    const float* hidden4  = (const float*)d_in[29];
    const float* last_hid = (const float*)d_in[30];
    const float* pat_tags = (const float*)d_in[31];
    // d_in[32] soft_label_llm unused in forward
    const int* category_id = (const int*)d_in[33];
    const int* source_id   = (const int*)d_in[34];
    const int* input_ids   = (const int*)d_in[35];
    const int* attn_mask   = (const int*)d_in[36];

    // output regions (return order: logits, z_base, router_w, router_l, aux)
    float* out      = (float*)d_out;
    float* o_logits = out;
    float* o_zb     = out + BB;
    float* o_rw     = o_zb + (long)BB * DBASE;
    float* o_rl     = o_rw + BB * KK;
    float* o_aux    = o_rl + BB * KK;

    // workspace layout
    char* ws = (char*)d_ws;
    int*   seg    = (int*)(ws + 0);                        // 128*4 ints
    float* pooled = (float*)(ws + 4096);                   // K*B*H
    float* h1     = pooled + (long)KK * BB * HH;           // K*B*H
    float* Z      = h1 + (long)KK * BB * HH;               // K*B*DEXP
    float* rin    = Z + (long)KK * BB * DEXP;              // B*DIN_PAD
    float* rh     = rin + (long)BB * DIN_PAD;              // B*512
    float* cin    = rh + (long)BB * 512;                   // B*DCLS_PAD
    float* c1     = cin + (long)BB * DCLS_PAD;             // B*1024
    float* c2     = c1 + (long)BB * 1024;                  // B*256

    seg_info_kernel<<<1, 128, 0, stream>>>(input_ids, attn_mask, seg);

    pool_kernel<<<dim3(3, BB), 256, 0, stream>>>(hidden4, last_hid, layer_w,
                                                 pat_tags, seg, o_zb);

    attn_kernel<<<BB, 256, 0, stream>>>(last_hid, attn_mask, attn_w, attn_b, pooled);

    // expert adapter layer 1: h1 = LN(gelu(pooled @ ad_w1 + ad_b1))
    gemm_wmma_kernel<<<dim3(HH / 64, BB / 32, KK), 256, 0, stream>>>(
        pooled, ad_w1, ad_b1, h1, BB, HH, HH, HH,
        (long)BB * HH, (long)HH * HH, (long)HH, (long)BB * HH, 1);
    ln_kernel<<<KK * BB, 256, 0, stream>>>(h1, ad_lng, ad_lnb, HH, HH, BB);

    // expert adapter layer 2: Z = h1 @ ad_w2 + ad_b2
    gemm_wmma_kernel<<<dim3(DEXP / 64, BB / 32, KK), 256, 0, stream>>>(
        h1, ad_w2, ad_b2, Z, BB, DEXP, HH, HH,
        (long)BB * HH, (long)HH * DEXP, (long)DEXP, (long)BB * DEXP, 0);

    aux_kernel<<<BB * KK, 32, 0, stream>>>(Z, aux_w, aux_b, o_aux);

    // router MLP
    rin_kernel<<<BB, 256, 0, stream>>>(o_zb, cat_emb, src_emb, pat_w, pat_b,
                                       pat_tags, category_id, source_id, rin);
    gemm_wmma_kernel<<<dim3(512 / 64, BB / 32, 1), 256, 0, stream>>>(
        rin, r_w1, r_b1, rh, BB, 512, DIN, DIN_PAD, 0, 0, 0, 0, 1);
    ln_kernel<<<BB, 256, 0, stream>>>(rh, r_lng, r_lnb, 512, 512, BB);
    router_kernel<<<BB, 32, 0, stream>>>(rh, r_w2, r_b2, o_rw, o_rl);

    // classifier
    cbuild_kernel<<<BB, 256, 0, stream>>>(o_zb, Z, o_rw, cin);
    ln_kernel<<<BB, 256, 0, stream>>>(cin, cls_lng, cls_lnb, DCLS, DCLS_PAD, BB);
    gemm_wmma_kernel<<<dim3(1024 / 64, BB / 32, 1), 256, 0, stream>>>(
        cin, cls_w1, cls_b1, c1, BB, 1024, DCLS, DCLS_PAD, 0, 0, 0, 0, 1);
    gemm_wmma_kernel<<<dim3(256 / 64, BB / 32, 1), 256, 0, stream>>>(
        c1, cls_w2, cls_b2, c2, BB, 256, 1024, 1024, 0, 0, 0, 0, 1);
    final_kernel<<<BB, 256, 0, stream>>>(c2, cls_w3, cls_b3, o_logits);
}